// RESEnTransformer_39848706573630
// MI455X (gfx1250) — compile-verified
//
#include <hip/hip_runtime.h>
#include <hip/hip_bf16.h>
#include <math.h>

// Problem constants (match reference)
static const int CB = 4;      // batch
static const int CN = 1024;   // tokens
static const int CD = 128;    // model dim
static const int CH_ = 8;     // heads
static const int CDH = 32;    // head dim
static const int CL = 4;      // layers
static const int CC = 20;     // classes
static const int CI = 256;    // H*DH
static const int CCH = 16;    // coor hidden
#define NEGV (-1000000000.0f)

typedef float v2f __attribute__((ext_vector_type(2)));
typedef float v8f __attribute__((ext_vector_type(8)));

__device__ __forceinline__ float gelu_exact(float x) {
  return 0.5f * x * (1.0f + erff(x * 0.70710678118654752440f));
}

__device__ __forceinline__ float wave_sum(float v) {
#pragma unroll
  for (int m = 16; m > 0; m >>= 1) v += __shfl_xor(v, m, 32);
  return v;
}
__device__ __forceinline__ float wave_max(float v) {
#pragma unroll
  for (int m = 16; m > 0; m >>= 1) v = fmaxf(v, __shfl_xor(v, m, 32));
  return v;
}

// ---------------- elementwise kernels ----------------
__global__ void k_embed(const int* __restrict__ tok, const float* __restrict__ emb,
                        float* __restrict__ feats, int total) {
  int idx = blockIdx.x * blockDim.x + threadIdx.x;
  if (idx >= total) return;
  int d = idx % CD;
  int r = idx / CD;
  feats[idx] = emb[tok[r] * CD + d];
}

__global__ void k_copy(float* __restrict__ dst, const float* __restrict__ src, int total) {
  int idx = blockIdx.x * blockDim.x + threadIdx.x;
  if (idx < total) dst[idx] = src[idx];
}

__global__ void k_add(float* __restrict__ dst, const float* __restrict__ src, int total) {
  int idx = blockIdx.x * blockDim.x + threadIdx.x;
  if (idx < total) dst[idx] += src[idx];
}

__global__ void k_bias_gelu(float* __restrict__ x, const float* __restrict__ bias,
                            int cols, int total) {
  int idx = blockIdx.x * blockDim.x + threadIdx.x;
  if (idx >= total) return;
  x[idx] = gelu_exact(x[idx] + bias[idx % cols]);
}

__global__ void k_residual_bias(float* __restrict__ feats, const float* __restrict__ src,
                                const float* __restrict__ bias, int cols, int total) {
  int idx = blockIdx.x * blockDim.x + threadIdx.x;
  if (idx >= total) return;
  feats[idx] += src[idx] + bias[idx % cols];
}

// LayerNorm: one wave32 per row of 128
__global__ __launch_bounds__(32) void k_layernorm(const float* __restrict__ x,
                                                  const float* __restrict__ g,
                                                  const float* __restrict__ b,
                                                  float* __restrict__ y) {
  int row = blockIdx.x;
  int lane = threadIdx.x;
  const float* xr = x + (size_t)row * CD;
  float v[4];
#pragma unroll
  for (int t = 0; t < 4; t++) v[t] = xr[lane + 32 * t];
  float mean = wave_sum(v[0] + v[1] + v[2] + v[3]) * (1.0f / CD);
  float var = 0.f;
#pragma unroll
  for (int t = 0; t < 4; t++) { float d = v[t] - mean; var += d * d; }
  var = wave_sum(var) * (1.0f / CD);
  float inv = rsqrtf(var + 1e-5f);
  float* yr = y + (size_t)row * CD;
#pragma unroll
  for (int t = 0; t < 4; t++) {
    int c = lane + 32 * t;
    yr[c] = (v[t] - mean) * inv * g[c] + b[c];
  }
}

// Rotary on q and k slices of qkv (B,N,3I) in-place
__global__ void k_rotary(float* __restrict__ qkv, int total) {
  int idx = blockIdx.x * blockDim.x + threadIdx.x;
  if (idx >= total) return;             // total = B*N*H*16
  int p = idx & 15;
  int t = idx >> 4;
  int h = t % CH_; t /= CH_;
  int n = t % CN;
  int b = t / CN;
  float inv = powf(10000.f, -((float)(2 * p)) / (float)CDH);
  float f = (float)n * inv;
  float c = cosf(f), s = sinf(f);
  size_t base = ((size_t)(b * CN + n)) * (3 * CI) + h * CDH + 2 * p;
  float x1 = qkv[base], x2 = qkv[base + 1];
  qkv[base] = x1 * c - x2 * s;
  qkv[base + 1] = x1 * s + x2 * c;
  x1 = qkv[base + CI]; x2 = qkv[base + CI + 1];
  qkv[base + CI] = x1 * c - x2 * s;
  qkv[base + CI + 1] = x1 * s + x2 * c;
}

// ---------------- WMMA f32 GEMM ----------------
// One wave computes a 16 x (16*NT) tile of C; the A fragment is loaded once
// per K-step and reused across NT WMMAs (cuts A traffic by NT).
// C[m,n] = alpha * sum_k A[m,k] * B[k,n]
// A: Ab[m*lda + k], B: Bb[k*rsB + n*csB], C: Cb[m*ldc + n]
// batch z = blockIdx.z decomposed as zo = z/Z2, zi = z%Z2 with separate strides.
template <int NT>
__global__ __launch_bounds__(32) void k_gemm_wmma(
    const float* __restrict__ A, const float* __restrict__ B, float* __restrict__ C,
    int K, int lda, long long sAo, long long sAi,
    int rsB, int csB, long long sBo, long long sBi,
    int ldc, long long sCo, long long sCi, int Z2, float alpha) {
  int z = blockIdx.z;
  int zo = z / Z2, zi = z % Z2;
  const float* Ab = A + zo * sAo + zi * sAi;
  const float* Bb = B + zo * sBo + zi * sBi;
  float* Cb = C + zo * sCo + zi * sCi;
  int lane = threadIdx.x;
  int half = lane >> 4;        // K-pair selector
  int l = lane & 15;
  int m0 = blockIdx.x * 16, n0 = blockIdx.y * (16 * NT);
  const float* Arow = Ab + (size_t)(m0 + l) * lda;
  const float* Bcol[NT];
#pragma unroll
  for (int t = 0; t < NT; t++) Bcol[t] = Bb + (size_t)(n0 + t * 16 + l) * csB;
  v8f acc[NT];
#pragma unroll
  for (int t = 0; t < NT; t++) acc[t] = (v8f){0.f, 0.f, 0.f, 0.f, 0.f, 0.f, 0.f, 0.f};
  for (int k = 0; k < K; k += 4) {
    int kb = k + 2 * half;
    v2f a;
    a.x = Arow[kb];
    a.y = Arow[kb + 1];
    __builtin_prefetch(Arow + kb + 16, 0, 1);   // global_prefetch_b8
#pragma unroll
    for (int t = 0; t < NT; t++) {
      v2f bfr;
      bfr.x = Bcol[t][(size_t)kb * rsB];
      bfr.y = Bcol[t][(size_t)(kb + 1) * rsB];
      acc[t] = __builtin_amdgcn_wmma_f32_16x16x4_f32(false, a, false, bfr,
                                                     (short)0, acc[t], false, false);
    }
  }
#pragma unroll
  for (int t = 0; t < NT; t++) {
#pragma unroll
    for (int r = 0; r < 8; r++) {
      int row = m0 + r + 8 * half;
      Cb[(size_t)row * ldc + n0 + t * 16 + l] = alpha * acc[t][r];
    }
  }
}

// ---------------- fused coordinate path ----------------
// For each (b,i): over j, read 8 head logits of scaled qk, run gelu(qk@W1)@W2,
// mask, weight normalized relative coords, accumulate rel_agg, contract with
// coors_combine -> delta (B,N,3). Never materializes cw (saves ~134MB/layer).
__global__ __launch_bounds__(256) void k_coor(
    const float* __restrict__ qk, const float* __restrict__ coords,
    const unsigned char* __restrict__ mask,
    const float* __restrict__ w1, const float* __restrict__ w2,
    const float* __restrict__ combine, const float* __restrict__ cscale_p,
    float* __restrict__ delta) {
  int bi = blockIdx.x;
  int b = bi / CN, i = bi % CN;
  __shared__ float sW1[CH_ * CCH];
  __shared__ float sW2[CCH * CH_];
  __shared__ float sred[8 * 24];
  int tid = threadIdx.x;
  if (tid < CH_ * CCH) sW1[tid] = w1[tid];
  else if (tid < 2 * CH_ * CCH) sW2[tid - CH_ * CCH] = w2[tid - CH_ * CCH];
  __syncthreads();
  float cs = cscale_p[0];
  const unsigned char* mrow = mask + b * CN;
  bool mi = mrow[i] != 0;
  float ci0 = coords[(size_t)(b * CN + i) * 3 + 0];
  float ci1 = coords[(size_t)(b * CN + i) * 3 + 1];
  float ci2 = coords[(size_t)(b * CN + i) * 3 + 2];
  float acc[24];
#pragma unroll
  for (int c = 0; c < 24; c++) acc[c] = 0.f;
  if (mi) {
    for (int j = tid; j < CN; j += 256) {
      if (!mrow[j]) continue;
      float qh[CH_];
#pragma unroll
      for (int h = 0; h < CH_; h++)
        qh[h] = qk[(((size_t)b * CH_ + h) * CN + i) * CN + j];
      float cw[CH_];
#pragma unroll
      for (int h = 0; h < CH_; h++) cw[h] = 0.f;
#pragma unroll
      for (int c = 0; c < CCH; c++) {
        float t = 0.f;
#pragma unroll
        for (int h = 0; h < CH_; h++) t += qh[h] * sW1[h * CCH + c];
        t = gelu_exact(t);
#pragma unroll
        for (int h = 0; h < CH_; h++) cw[h] += t * sW2[c * CH_ + h];
      }
      float r0 = ci0 - coords[(size_t)(b * CN + j) * 3 + 0];
      float r1 = ci1 - coords[(size_t)(b * CN + j) * 3 + 1];
      float r2 = ci2 - coords[(size_t)(b * CN + j) * 3 + 2];
      float nrm = sqrtf(r0 * r0 + r1 * r1 + r2 * r2 + 1e-12f);
      float invn = cs / fmaxf(nrm, 1e-8f);
      r0 *= invn; r1 *= invn; r2 *= invn;
#pragma unroll
      for (int h = 0; h < CH_; h++) {
        acc[h * 3 + 0] += cw[h] * r0;
        acc[h * 3 + 1] += cw[h] * r1;
        acc[h * 3 + 2] += cw[h] * r2;
      }
    }
  }
  int lane = tid & 31, w = tid >> 5;
#pragma unroll
  for (int c = 0; c < 24; c++) {
    float v = wave_sum(acc[c]);
    if (lane == 0) sred[w * 24 + c] = v;
  }
  __syncthreads();
  if (tid == 0) {
    float o0 = 0.f, o1 = 0.f, o2 = 0.f;
#pragma unroll
    for (int h = 0; h < CH_; h++) {
      float a0 = 0.f, a1 = 0.f, a2 = 0.f;
      for (int ww = 0; ww < 8; ww++) {
        a0 += sred[ww * 24 + h * 3 + 0];
        a1 += sred[ww * 24 + h * 3 + 1];
        a2 += sred[ww * 24 + h * 3 + 2];
      }
      float cb = combine[h];
      o0 += a0 * cb; o1 += a1 * cb; o2 += a2 * cb;
    }
    delta[(size_t)(b * CN + i) * 3 + 0] = o0;
    delta[(size_t)(b * CN + i) * 3 + 1] = o1;
    delta[(size_t)(b * CN + i) * 3 + 2] = o2;
  }
}

// masked softmax over j, in-place on qk rows (already scaled)
__global__ __launch_bounds__(256) void k_softmax(float* __restrict__ qk,
                                                 const unsigned char* __restrict__ mask) {
  int row = blockIdx.x;                 // (b*H + h)*N + i
  int i = row & (CN - 1);
  int b = row / (CH_ * CN);
  float* p = qk + (size_t)row * CN;
  const unsigned char* mrow = mask + b * CN;
  bool mi = mrow[i] != 0;
  int tid = threadIdx.x;
  float vals[4];
  float vmax = -3.0e38f;
#pragma unroll
  for (int t = 0; t < 4; t++) {
    int j = tid + 256 * t;
    float v = (mi && mrow[j]) ? p[j] : NEGV;
    vals[t] = v;
    vmax = fmaxf(vmax, v);
  }
  __shared__ float sred[8];
  int lane = tid & 31, w = tid >> 5;
  vmax = wave_max(vmax);
  if (lane == 0) sred[w] = vmax;
  __syncthreads();
  float bmax = sred[0];
#pragma unroll
  for (int k = 1; k < 8; k++) bmax = fmaxf(bmax, sred[k]);
  __syncthreads();
  float sum = 0.f;
#pragma unroll
  for (int t = 0; t < 4; t++) {
    float e = __expf(vals[t] - bmax);
    vals[t] = e;
    sum += e;
  }
  sum = wave_sum(sum);
  if (lane == 0) sred[w] = sum;
  __syncthreads();
  float bs = 0.f;
#pragma unroll
  for (int k = 0; k < 8; k++) bs += sred[k];
  float invs = 1.0f / bs;
#pragma unroll
  for (int t = 0; t < 4; t++) p[tid + 256 * t] = vals[t] * invs;
}

// final head: central token -> LN -> 128->256 gelu -> 256->20
__global__ __launch_bounds__(256) void k_head(
    const float* __restrict__ feats, const float* __restrict__ g,
    const float* __restrict__ bta, const float* __restrict__ w1,
    const float* __restrict__ b1, const float* __restrict__ w2,
    const float* __restrict__ b2, float* __restrict__ out) {
  __shared__ float buf[CD];
  __shared__ float red[64];
  __shared__ float hid[CB][2 * CD];
  int tid = threadIdx.x;
  for (int b = 0; b < CB; b++) {
    if (tid < CD) buf[tid] = feats[((size_t)b * CN) * CD + tid];
    __syncthreads();
    if (tid < 64) red[tid] = buf[tid] + buf[tid + 64];
    __syncthreads();
    for (int s = 32; s > 0; s >>= 1) {
      if (tid < s) red[tid] += red[tid + s];
      __syncthreads();
    }
    float mean = red[0] * (1.0f / CD);
    __syncthreads();
    if (tid < 64) {
      float a = buf[tid] - mean, c = buf[tid + 64] - mean;
      red[tid] = a * a + c * c;
    }
    __syncthreads();
    for (int s = 32; s > 0; s >>= 1) {
      if (tid < s) red[tid] += red[tid + s];
      __syncthreads();
    }
    float inv = rsqrtf(red[0] * (1.0f / CD) + 1e-5f);
    __syncthreads();
    if (tid < CD) buf[tid] = (buf[tid] - mean) * inv * g[tid] + bta[tid];
    __syncthreads();
    float a = b1[tid];
    for (int d = 0; d < CD; d++) a += buf[d] * w1[d * (2 * CD) + tid];
    hid[b][tid] = gelu_exact(a);
    __syncthreads();
  }
  if (tid < CB * CC) {
    int b = tid / CC, c = tid % CC;
    float a = b2[c];
    for (int k = 0; k < 2 * CD; k++) a += hid[b][k] * w2[k * CC + c];
    out[b * CC + c] = a;
  }
}

// ---------------- host ----------------
static inline void launch_gemm(const float* A, const float* B, float* C,
                               int M, int N, int K,
                               int lda, long long sAo, long long sAi,
                               int rsB, int csB, long long sBo, long long sBi,
                               int ldc, long long sCo, long long sCi,
                               int Z2, int batches, float alpha, int ntiles,
                               hipStream_t s) {
  if (ntiles == 4) {
    dim3 grid(M / 16, N / 64, batches);
    k_gemm_wmma<4><<<grid, 32, 0, s>>>(A, B, C, K, lda, sAo, sAi, rsB, csB, sBo,
                                       sBi, ldc, sCo, sCi, Z2, alpha);
  } else if (ntiles == 2) {
    dim3 grid(M / 16, N / 32, batches);
    k_gemm_wmma<2><<<grid, 32, 0, s>>>(A, B, C, K, lda, sAo, sAi, rsB, csB, sBo,
                                       sBi, ldc, sCo, sCi, Z2, alpha);
  } else {
    dim3 grid(M / 16, N / 16, batches);
    k_gemm_wmma<1><<<grid, 32, 0, s>>>(A, B, C, K, lda, sAo, sAi, rsB, csB, sBo,
                                       sBi, ldc, sCo, sCi, Z2, alpha);
  }
}

extern "C" void kernel_launch(void* const* d_in, const int* in_sizes, int n_in,
                              void* d_out, int out_size, void* d_ws, size_t ws_size,
                              hipStream_t stream) {
  (void)in_sizes; (void)n_in; (void)out_size; (void)ws_size;
  const int* atom = (const int*)d_in[0];
  const float* coords_in = (const float*)d_in[1];
  const unsigned char* mask = (const unsigned char*)d_in[2];
  const float* tok_emb = (const float*)d_in[3];
  const float* attn_ln_g = (const float*)d_in[4];
  const float* attn_ln_b = (const float*)d_in[5];
  const float* w_qkv = (const float*)d_in[6];
  const float* w_out = (const float*)d_in[7];
  const float* b_out = (const float*)d_in[8];
  const float* coor_w1 = (const float*)d_in[9];
  const float* coor_w2 = (const float*)d_in[10];
  const float* coors_combine = (const float*)d_in[11];
  const float* coors_scale = (const float*)d_in[12];
  const float* ff_ln_g = (const float*)d_in[13];
  const float* ff_ln_b = (const float*)d_in[14];
  const float* ff_w1 = (const float*)d_in[15];
  const float* ff_b1 = (const float*)d_in[16];
  const float* ff_w2 = (const float*)d_in[17];
  const float* ff_b2 = (const float*)d_in[18];
  const float* head_ln_g = (const float*)d_in[19];
  const float* head_ln_b = (const float*)d_in[20];
  const float* head_w1 = (const float*)d_in[21];
  const float* head_b1 = (const float*)d_in[22];
  const float* head_w2 = (const float*)d_in[23];
  const float* head_b2 = (const float*)d_in[24];

  float* ws = (float*)d_ws;
  const size_t nRows = (size_t)CB * CN;          // 4096
  float* feats = ws;                              // 4096*128
  float* xbuf = feats + nRows * CD;               // 4096*128
  float* qkv = xbuf + nRows * CD;                 // 4096*768
  float* qkbuf = qkv + nRows * 3 * CI;            // 4*8*1024*1024
  float* outc = qkbuf + (size_t)CB * CH_ * CN * CN;  // 4096*256
  float* proj = outc + nRows * CI;                // 4096*128
  float* ff1 = proj + nRows * CD;                 // 4096*512
  float* ff2 = ff1 + nRows * 4 * CD;              // 4096*128
  float* coords_cur = ff2 + nRows * CD;           // 4096*3
  float* delta = coords_cur + nRows * 3;          // 4096*3

  const float scale = 0.17677669529663687f;       // 1/sqrt(32)
  const int TPB = 256;

  k_embed<<<((int)(nRows * CD) + TPB - 1) / TPB, TPB, 0, stream>>>(atom, tok_emb, feats,
                                                                   (int)(nRows * CD));
  k_copy<<<((int)(nRows * 3) + TPB - 1) / TPB, TPB, 0, stream>>>(coords_cur, coords_in,
                                                                 (int)(nRows * 3));

  for (int l = 0; l < CL; l++) {
    // --- attention block ---
    k_layernorm<<<(int)nRows, 32, 0, stream>>>(feats, attn_ln_g + l * CD,
                                               attn_ln_b + l * CD, xbuf);
    // QKV: (4096x128) @ (128x768)
    launch_gemm(xbuf, w_qkv + (size_t)l * CD * 3 * CI, qkv, (int)nRows, 3 * CI, CD,
                CD, 0, 0, 3 * CI, 1, 0, 0, 3 * CI, 0, 0, 1, 1, 1.0f, 4, stream);
    k_rotary<<<((int)(nRows * CH_ * 16) + TPB - 1) / TPB, TPB, 0, stream>>>(
        qkv, (int)(nRows * CH_ * 16));
    // qk[b,h,i,j] = scale * q . k  : 32 batched 1024x32x1024 GEMMs
    launch_gemm(qkv, qkv + CI, qkbuf, CN, CN, CDH,
                3 * CI, (long long)CN * 3 * CI, CDH,
                1, 3 * CI, (long long)CN * 3 * CI, CDH,
                CN, (long long)CH_ * CN * CN, (long long)CN * CN,
                CH_, CB * CH_, scale, 4, stream);
    // coordinate path (reads scaled qk pre-mask)
    k_coor<<<(int)nRows, 256, 0, stream>>>(qkbuf, coords_cur, mask,
                                           coor_w1 + l * CH_ * CCH,
                                           coor_w2 + l * CCH * CH_,
                                           coors_combine + l * CH_,
                                           coors_scale + l, delta);
    // masked softmax in place
    k_softmax<<<CB * CH_ * CN, 256, 0, stream>>>(qkbuf, mask);
    // out[b,i,h*32+d] = attn @ v : 32 batched 1024x1024x32 GEMMs
    launch_gemm(qkbuf, qkv + 2 * CI, outc, CN, CDH, CN,
                CN, (long long)CH_ * CN * CN, (long long)CN * CN,
                3 * CI, 1, (long long)CN * 3 * CI, CDH,
                CI, (long long)CN * CI, CDH,
                CH_, CB * CH_, 1.0f, 2, stream);
    // output projection (4096x256)@(256x128)
    launch_gemm(outc, w_out + (size_t)l * CI * CD, proj, (int)nRows, CD, CI,
                CI, 0, 0, CD, 1, 0, 0, CD, 0, 0, 1, 1, 1.0f, 4, stream);
    k_residual_bias<<<((int)(nRows * CD) + TPB - 1) / TPB, TPB, 0, stream>>>(
        feats, proj, b_out + l * CD, CD, (int)(nRows * CD));
    k_add<<<((int)(nRows * 3) + TPB - 1) / TPB, TPB, 0, stream>>>(coords_cur, delta,
                                                                  (int)(nRows * 3));
    // --- FFN ---
    k_layernorm<<<(int)nRows, 32, 0, stream>>>(feats, ff_ln_g + l * CD,
                                               ff_ln_b + l * CD, xbuf);
    launch_gemm(xbuf, ff_w1 + (size_t)l * CD * 4 * CD, ff1, (int)nRows, 4 * CD, CD,
                CD, 0, 0, 4 * CD, 1, 0, 0, 4 * CD, 0, 0, 1, 1, 1.0f, 4, stream);
    k_bias_gelu<<<((int)(nRows * 4 * CD) + TPB - 1) / TPB, TPB, 0, stream>>>(
        ff1, ff_b1 + l * 4 * CD, 4 * CD, (int)(nRows * 4 * CD));
    launch_gemm(ff1, ff_w2 + (size_t)l * 4 * CD * CD, ff2, (int)nRows, CD, 4 * CD,
                4 * CD, 0, 0, CD, 1, 0, 0, CD, 0, 0, 1, 1, 1.0f, 4, stream);
    k_residual_bias<<<((int)(nRows * CD) + TPB - 1) / TPB, TPB, 0, stream>>>(
        feats, ff2, ff_b2 + l * CD, CD, (int)(nRows * CD));
  }

  k_head<<<1, 256, 0, stream>>>(feats, head_ln_g, head_ln_b, head_w1, head_b1,
                                head_w2, head_b2, (float*)d_out);
}